// NGCFLayer_4063039062696
// MI455X (gfx1250) — compile-verified
//
#include <hip/hip_runtime.h>
#include <hip/hip_bf16.h>

// NGCF layer for MI455X (gfx1250, wave32).
// Precision: fp32 throughout (matches reference) using V_WMMA_F32_16X16X4_F32.
//
// Pipeline (all on `stream`):
//   memset(macc=0)
//   k1: h1 = h @ W1^T                 (WMMA, 16 nodes / wave)
//   k2: per-edge msg + atomic scatter (WMMA, 16 edges / wave)
//   k3: m+h1 -> LeakyReLU -> L2 norm  (1 wave / node)

typedef float v2f __attribute__((ext_vector_type(2)));
typedef float v8f __attribute__((ext_vector_type(8)));

#define NUSERS 60000
#define NITEMS 40000
#define NNODES 100000   // NUSERS + NITEMS, divisible by 16
#define NEDGES 1600000  // divisible by 16
#define DIM 64
#define LDSW 68         // LDS row stride (floats): 68 mod 64 = 4 -> the 16 lanes'
                        // float2 B reads tile all 64 banks conflict-free.

__device__ __forceinline__ const float* node_row(const float* __restrict__ U,
                                                 const float* __restrict__ I, int n) {
  return (n < NUSERS) ? (U + (size_t)n * DIM) : (I + (size_t)(n - NUSERS) * DIM);
}

// Cooperative load of a 64x64 row-major weight into bank-padded LDS.
__device__ __forceinline__ void load_w_lds(float* __restrict__ w,
                                           const float* __restrict__ W) {
  for (int idx = threadIdx.x; idx < DIM * DIM; idx += 256)
    w[(idx >> 6) * LDSW + (idx & 63)] = W[idx];
  __syncthreads();
}

// ---------------------------------------------------------------------------
// Kernel 1: h1[n, :] = h[n, :] @ W1^T   (h1[n,o] = sum_k h[n,k] * W1[o,k])
// One wave handles a 16-node x 64-output tile: 4 N-tiles x 16 K-steps of
// v_wmma_f32_16x16x4_f32.
// ---------------------------------------------------------------------------
__global__ void __launch_bounds__(256)
node_transform_kernel(const float* __restrict__ U, const float* __restrict__ I,
                      const float* __restrict__ W1, float* __restrict__ h1) {
  __shared__ float w[DIM * LDSW];
  load_w_lds(w, W1);

  const int wave = threadIdx.x >> 5;
  const int lane = threadIdx.x & 31;
  const int tile = blockIdx.x * 8 + wave;
  if (tile >= NNODES / 16) return;  // wave-uniform: EXEC stays all-1s for WMMA

  const int m = lane & 15;      // A row / B col within tile
  const int half = lane >> 4;   // selects K pair {0,1} vs {2,3} per step

  const float* hrow = node_row(U, I, tile * 16 + m);

  // A operand, ISA 32-bit 16x4 layout: av[k] = {A[m][4k+2h], A[m][4k+2h+1]}
  v2f av[16];
#pragma unroll
  for (int k = 0; k < 16; ++k)
    av[k] = *(const v2f*)(hrow + 4 * k + 2 * half);

#pragma unroll
  for (int nt = 0; nt < 4; ++nt) {
    v8f acc = {};
#pragma unroll
    for (int k = 0; k < 16; ++k) {
      // B[k][n] = W1[o][kg]: b = {W1[o][4k+2h], W1[o][4k+2h+1]}, o = nt*16+m
      v2f b = *(const v2f*)(&w[(nt * 16 + m) * LDSW + 4 * k + 2 * half]);
      acc = __builtin_amdgcn_wmma_f32_16x16x4_f32(false, av[k], false, b,
                                                  (short)0, acc, false, false);
    }
    // D layout: VGPR j holds row M = j + 8*half, col N = lane&15
    const int o = nt * 16 + m;
#pragma unroll
    for (int j = 0; j < 8; ++j)
      h1[(size_t)(tile * 16 + j + 8 * half) * DIM + o] = acc[j];
  }
}

// ---------------------------------------------------------------------------
// Kernel 2: per-edge messages. One wave = 16 edges.
//   A[m][k] = h[src[e0+m]][k] * h[dst[e0+m]][k]   (gathered as float2 pairs,
//             already in WMMA A layout), B = W2 from LDS.
//   msg = (h1[src] + A@W2^T) * norm[src]*norm[dst]; atomicAdd into macc[dst].
// ---------------------------------------------------------------------------
__global__ void __launch_bounds__(256)
edge_message_kernel(const float* __restrict__ U, const float* __restrict__ I,
                    const float* __restrict__ norm, const int* __restrict__ src,
                    const int* __restrict__ dst, const float* __restrict__ W2,
                    const float* __restrict__ h1, float* __restrict__ macc) {
  __shared__ float w[DIM * LDSW];
  load_w_lds(w, W2);

  const int wave = threadIdx.x >> 5;
  const int lane = threadIdx.x & 31;
  const int tile = blockIdx.x * 8 + wave;  // grid sized exactly: no guard needed
  const int e0 = tile * 16;

  const int m = lane & 15;
  const int half = lane >> 4;

  // Gather the two endpoint rows for A-row m; both halves read the same rows
  // but disjoint column pairs (kg = 4k + 2*half).
  const int sA = src[e0 + m];
  const int dA = dst[e0 + m];
  const float* hs = node_row(U, I, sA);
  const float* hd = node_row(U, I, dA);

  v2f av[16];
#pragma unroll
  for (int k = 0; k < 16; ++k) {
    const int kg = 4 * k + 2 * half;
    v2f a = *(const v2f*)(hs + kg);
    v2f b = *(const v2f*)(hd + kg);
    av[k] = a * b;  // element-wise h_src * h_dst, already in A layout
  }

  // Epilogue metadata: this lane's 8 output rows are edges j + 8*half.
  int sj[8], dj[8];
  float wj[8];
#pragma unroll
  for (int j = 0; j < 8; ++j) {
    const int e = e0 + j + 8 * half;
    sj[j] = src[e];
    dj[j] = dst[e];
    wj[j] = norm[sj[j]] * norm[dj[j]];
  }

#pragma unroll
  for (int nt = 0; nt < 4; ++nt) {
    v8f acc = {};
#pragma unroll
    for (int k = 0; k < 16; ++k) {
      v2f b = *(const v2f*)(&w[(nt * 16 + m) * LDSW + 4 * k + 2 * half]);
      acc = __builtin_amdgcn_wmma_f32_16x16x4_f32(false, av[k], false, b,
                                                  (short)0, acc, false, false);
    }
    const int o = nt * 16 + m;
#pragma unroll
    for (int j = 0; j < 8; ++j) {
      const float val = (h1[(size_t)sj[j] * DIM + o] + acc[j]) * wj[j];
      unsafeAtomicAdd(&macc[(size_t)dj[j] * DIM + o], val);  // global_atomic_add_f32
    }
  }
}

// ---------------------------------------------------------------------------
// Kernel 3: out = l2norm(leakyrelu(macc + h1)). One wave per node row.
// ---------------------------------------------------------------------------
__global__ void __launch_bounds__(256)
finalize_kernel(const float* __restrict__ h1, const float* __restrict__ macc,
                float* __restrict__ out) {
  const int wave = threadIdx.x >> 5;
  const int lane = threadIdx.x & 31;
  const int node = blockIdx.x * 8 + wave;
  if (node >= NNODES) return;

  const size_t base = (size_t)node * DIM;
  float v0 = macc[base + lane] + h1[base + lane];
  float v1 = macc[base + lane + 32] + h1[base + lane + 32];
  v0 = (v0 >= 0.0f) ? v0 : 0.2f * v0;
  v1 = (v1 >= 0.0f) ? v1 : 0.2f * v1;

  float ss = v0 * v0 + v1 * v1;
#pragma unroll
  for (int off = 16; off > 0; off >>= 1) ss += __shfl_xor(ss, off, 32);

  const float inv = 1.0f / fmaxf(sqrtf(ss), 1e-12f);
  out[base + lane] = v0 * inv;
  out[base + lane + 32] = v1 * inv;
}

// ---------------------------------------------------------------------------
extern "C" void kernel_launch(void* const* d_in, const int* in_sizes, int n_in,
                              void* d_out, int out_size, void* d_ws, size_t ws_size,
                              hipStream_t stream) {
  const float* U    = (const float*)d_in[0];  // [60000, 64]
  const float* I    = (const float*)d_in[1];  // [40000, 64]
  const float* norm = (const float*)d_in[2];  // [100000, 1]
  const int*   src  = (const int*)d_in[3];    // [1600000]
  const int*   dst  = (const int*)d_in[4];    // [1600000]
  const float* W1   = (const float*)d_in[5];  // [64, 64]
  const float* W2   = (const float*)d_in[6];  // [64, 64]
  float* out = (float*)d_out;

  float* h1   = (float*)d_ws;                  // 100000*64 f32 = 25.6 MB
  float* macc = h1 + (size_t)NNODES * DIM;     // 100000*64 f32 = 25.6 MB

  hipMemsetAsync(macc, 0, (size_t)NNODES * DIM * sizeof(float), stream);

  node_transform_kernel<<<(NNODES / 16 + 7) / 8, 256, 0, stream>>>(U, I, W1, h1);
  edge_message_kernel<<<(NEDGES / 16) / 8, 256, 0, stream>>>(U, I, norm, src, dst,
                                                             W2, h1, macc);
  finalize_kernel<<<(NNODES + 7) / 8, 256, 0, stream>>>(h1, macc, out);
}